// ClarityTexture_8315056685430
// MI455X (gfx1250) — compile-verified
//
#include <hip/hip_runtime.h>
#include <cmath>

// Problem constants
#define HQ 1024
#define WQ 1024
#define HW (HQ * WQ)          // 1048576 pixels per plane
#define NB 8                  // batch
#define NC 3                  // channels

typedef __attribute__((ext_vector_type(2))) float v2f;
typedef __attribute__((ext_vector_type(8))) float v8f;

struct GW { float g31[31]; float g7[7]; };

__device__ __forceinline__ int iclamp(int v, int lo, int hi) {
    return v < lo ? lo : (v > hi ? hi : v);
}

// ---------------------------------------------------------------------------
// Pass 1: luma = 0.2126 R + 0.7152 G + 0.0722 B   (float4 vectorized, B128)
// ---------------------------------------------------------------------------
__global__ __launch_bounds__(256) void luma_kernel(const float* __restrict__ x,
                                                   float* __restrict__ L) {
    int i = blockIdx.x * 256 + threadIdx.x;       // over NB*HW/4 float4 groups
    int b   = i >> 18;                            // HW/4 == 262144 == 1<<18
    int rem = i & ((HW / 4) - 1);
    const float4* xv = (const float4*)x;
    float4 r  = xv[(size_t)(b * 3 + 0) * (HW / 4) + rem];
    float4 g  = xv[(size_t)(b * 3 + 1) * (HW / 4) + rem];
    float4 bl = xv[(size_t)(b * 3 + 2) * (HW / 4) + rem];
    float4 o;
    o.x = 0.2126f * r.x + 0.7152f * g.x + 0.0722f * bl.x;
    o.y = 0.2126f * r.y + 0.7152f * g.y + 0.0722f * bl.y;
    o.z = 0.2126f * r.z + 0.7152f * g.z + 0.0722f * bl.z;
    o.w = 0.2126f * r.w + 0.7152f * g.w + 0.0722f * bl.w;
    ((float4*)L)[(size_t)b * (HW / 4) + rem] = o;
}

// ---------------------------------------------------------------------------
// Pass 2: vertical blurs (31-tap and 7-tap), zero padding. Coalesced along x;
// all tap rows are L2-resident (luma plane = 33 MB << 192 MB L2).
// ---------------------------------------------------------------------------
__global__ __launch_bounds__(256) void vblur_kernel(const float* __restrict__ L,
                                                    float* __restrict__ vb31,
                                                    float* __restrict__ vb7,
                                                    GW gw) {
    int i = blockIdx.x * 256 + threadIdx.x;       // over NB*HW pixels
    int b  = i >> 20;
    int p  = i & (HW - 1);
    int y  = p >> 10;
    int xx = p & (WQ - 1);
    const float* base = L + (size_t)b * HW + xx;

    float a31 = 0.f;
#pragma unroll
    for (int t = 0; t < 31; ++t) {
        int yy = y + t - 15;
        int yc = iclamp(yy, 0, HQ - 1);
        float m = (yy == yc) ? 1.f : 0.f;         // zero padding
        a31 = fmaf(gw.g31[t], base[(size_t)yc * WQ] * m, a31);
    }
    float a7 = 0.f;
#pragma unroll
    for (int t = 0; t < 7; ++t) {
        int yy = y + t - 3;
        int yc = iclamp(yy, 0, HQ - 1);
        float m = (yy == yc) ? 1.f : 0.f;
        a7 = fmaf(gw.g7[t], base[(size_t)yc * WQ] * m, a7);
    }
    vb31[i] = a31;
    vb7[i]  = a7;
}

// ---------------------------------------------------------------------------
// Pass 3: horizontal blurs as WMMA f32 matmuls against Toeplitz band + combine.
// One wave (32 lanes) per 16x16 output tile.
//   out[m][n] = sum_K in[m][K] * T[K][n],  T[K][n] = g[K-n] (0 <= K-n < taps)
//   31-tap: K window 46 -> 12 x V_WMMA_F32_16X16X4_F32 (full fp32, exact)
//    7-tap: K window 22 ->  6 x V_WMMA_F32_16X16X4_F32
// Toeplitz fragments depend only on (lane, step): precomputed branchlessly
// into VGPRs once per wave, so the WMMA loop is pure load-A -> wmma.
// ---------------------------------------------------------------------------
__global__ __launch_bounds__(32) void hblur_combine_kernel(
    const float* __restrict__ x, const float* __restrict__ L,
    const float* __restrict__ VB31, const float* __restrict__ VB7,
    const float* __restrict__ pclar, const float* __restrict__ ptex,
    float* __restrict__ out) {
    __shared__ float sg31[31];
    __shared__ float sg7[7];

    const int lane = threadIdx.x;                 // 0..31
    const int ln   = lane & 15;                   // N (and A's M) index
    const int hi   = lane >> 4;                   // half-wave K offset

    // Build normalized gaussian taps in LDS (fp32, matches reference's np path)
    if (lane < 31) { float d = (float)lane - 15.f; sg31[lane] = expf(-d * d * (1.f / 128.f)); }
    if (lane < 7)  { float d = (float)lane - 3.f;  sg7[lane]  = expf(-d * d * (1.f / 4.5f)); }
    __syncthreads();
    float s31 = 0.f, s7 = 0.f;
    for (int t = 0; t < 31; ++t) s31 += sg31[t];
    for (int t = 0; t < 7;  ++t) s7  += sg7[t];
    const float inv31 = 1.f / s31, inv7 = 1.f / s7;

    // Precompute Toeplitz B fragments for every K-step (branchless: clamp the
    // tap index so the LDS read is unconditional, mask the value by validity).
    v2f w31[12];
#pragma unroll
    for (int s = 0; s < 12; ++s) {
#pragma unroll
        for (int v = 0; v < 2; ++v) {
            int t  = 4 * s + 2 * hi + v - ln;     // T[K][n] = g[K-n]
            int tc = iclamp(t, 0, 30);
            w31[s][v] = sg31[tc] * ((t == tc) ? inv31 : 0.f);
        }
    }
    v2f w7[6];
#pragma unroll
    for (int s = 0; s < 6; ++s) {
#pragma unroll
        for (int v = 0; v < 2; ++v) {
            int t  = 4 * s + 2 * hi + v - ln;
            int tc = iclamp(t, 0, 6);
            w7[s][v] = sg7[tc] * ((t == tc) ? inv7 : 0.f);
        }
    }

    const int x0 = blockIdx.x * 16;
    const int y0 = blockIdx.y * 16;
    const int b  = blockIdx.z;
    const int row = y0 + ln;                      // A-fragment row (M = lane%16)
    const float* p31 = VB31 + (size_t)b * HW + (size_t)row * WQ;
    const float* p7  = VB7  + (size_t)b * HW + (size_t)row * WQ;
    const int kbase = 2 * hi;

    // 31-tap horizontal blur via chained fp32 WMMA
    v8f acc31 = {0.f, 0.f, 0.f, 0.f, 0.f, 0.f, 0.f, 0.f};
#pragma unroll
    for (int s = 0; s < 12; ++s) {
        v2f a;
#pragma unroll
        for (int v = 0; v < 2; ++v) {
            int col = x0 - 15 + 4 * s + kbase + v;  // window starts at x0 - 15
            int cc  = iclamp(col, 0, WQ - 1);
            a[v] = p31[cc] * ((col == cc) ? 1.f : 0.f);
        }
        acc31 = __builtin_amdgcn_wmma_f32_16x16x4_f32(
            false, a, false, w31[s], (short)0, acc31, false, false);
    }

    // 7-tap horizontal blur via chained fp32 WMMA
    v8f acc7 = {0.f, 0.f, 0.f, 0.f, 0.f, 0.f, 0.f, 0.f};
#pragma unroll
    for (int s = 0; s < 6; ++s) {
        v2f a;
#pragma unroll
        for (int v = 0; v < 2; ++v) {
            int col = x0 - 3 + 4 * s + kbase + v;   // window starts at x0 - 3
            int cc  = iclamp(col, 0, WQ - 1);
            a[v] = p7[cc] * ((col == cc) ? 1.f : 0.f);
        }
        acc7 = __builtin_amdgcn_wmma_f32_16x16x4_f32(
            false, a, false, w7[s], (short)0, acc7, false, false);
    }

    // Combine + ratio-scale 3 channels. D layout: lane -> N=ln, M=r+8*hi.
    const float ca = tanhf(pclar[0]) * 0.5f;
    const float ta = tanhf(ptex[0])  * 0.3f;
#pragma unroll
    for (int r = 0; r < 8; ++r) {
        int y  = y0 + r + 8 * hi;
        int xc = x0 + ln;
        size_t pidx = (size_t)b * HW + (size_t)y * WQ + xc;
        float Lv  = L[pidx];
        float le  = Lv + ca * (Lv - acc31[r]) + ta * (Lv - acc7[r]);
        float ratio = (le + 1e-6f) / (Lv + 1e-6f);
#pragma unroll
        for (int c = 0; c < 3; ++c) {
            size_t xi = (size_t)(b * 3 + c) * HW + (size_t)y * WQ + xc;
            float vv = x[xi] * ratio;
            out[xi] = fminf(fmaxf(vv, 0.f), 1.f);
        }
    }
}

// ---------------------------------------------------------------------------
extern "C" void kernel_launch(void* const* d_in, const int* in_sizes, int n_in,
                              void* d_out, int out_size, void* d_ws, size_t ws_size,
                              hipStream_t stream) {
    const float* x    = (const float*)d_in[0];    // (8,3,1024,1024) f32
    const float* clar = (const float*)d_in[1];    // scalar
    const float* tex  = (const float*)d_in[2];    // scalar
    float* out = (float*)d_out;

    // Workspace layout: luma | vblur31 | vblur7  (24 * 2^20 floats = 96 MB)
    float* ws   = (float*)d_ws;
    float* L    = ws;
    float* VB31 = ws + (size_t)NB * HW;
    float* VB7  = ws + (size_t)2 * NB * HW;

    // Host-computed gaussian taps (deterministic), passed by value (kernarg).
    GW gw;
    {
        float s = 0.f;
        for (int i = 0; i < 31; ++i) { float d = (float)(i - 15); gw.g31[i] = expf(-d * d / (2.f * 8.f * 8.f)); s += gw.g31[i]; }
        for (int i = 0; i < 31; ++i) gw.g31[i] /= s;
        s = 0.f;
        for (int i = 0; i < 7; ++i)  { float d = (float)(i - 3);  gw.g7[i]  = expf(-d * d / (2.f * 1.5f * 1.5f)); s += gw.g7[i]; }
        for (int i = 0; i < 7; ++i)  gw.g7[i] /= s;
    }

    // Pass 1: luma (NB*HW/4 float4 groups)
    luma_kernel<<<(NB * HW / 4) / 256, 256, 0, stream>>>(x, L);
    // Pass 2: vertical blurs (NB*HW pixels)
    vblur_kernel<<<(NB * HW) / 256, 256, 0, stream>>>(L, VB31, VB7, gw);
    // Pass 3: horizontal blurs via WMMA + combine (one wave per 16x16 tile)
    dim3 grid3(WQ / 16, HQ / 16, NB);
    hblur_combine_kernel<<<grid3, 32, 0, stream>>>(x, L, VB31, VB7, clar, tex, out);
}